// MimicNetLSTM_23545010717534
// MI455X (gfx1250) — compile-verified
//
#include <hip/hip_runtime.h>
#include <math.h>

// MimicNet 2-layer LSTM, batch=1, seq=1.  I=512, H=4096, gate rows R=4H=16384.
// Memory-bound (~840MB f32 weights @ 23.3TB/s => ~36us floor).  GEMV via
// V_WMMA_F32_16X16X4_F32: one wave owns 16 rows; the input vector occupies
// column 0 of B, so column 0 of D accumulates the 16 row-dot-products.
// Split-K x8 for occupancy (8192 waves), deterministic partial-sum reduction
// (fixed-order sums, no FP atomics).

#define IN_SZ 512
#define H_SZ 4096
#define R_SZ (4 * H_SZ) /* 16384 gate rows */
#define SPLIT 8

typedef __attribute__((ext_vector_type(2))) float v2f;
typedef __attribute__((ext_vector_type(8))) float v8f;

__device__ __forceinline__ float sigmoidf_(float x) {
  return 1.0f / (1.0f + expf(-x));
}

// D = A(16x4) * B(4x16) + C.
// A layout (ISA 7.12.2, 32-bit A 16x4): lane m (0-15) VGPR0/1 = W[row m][k0..k0+1];
// lane m+16 VGPR0/1 = W[row m][k0+2..3]  -> one float2 load per lane.
// B: only column N=0 populated: lane0 = {x[k0],x[k0+1]}, lane16 = {x[k0+2],x[k0+3]}.
// D column 0: lane0 VGPR m = row m, lane16 VGPR m = row m+8.
__device__ __forceinline__ v8f wmma_step(v8f c, v2f a, v2f b) {
  return __builtin_amdgcn_wmma_f32_16x16x4_f32(
      /*neg_a=*/false, a, /*neg_b=*/false, b,
      /*c_mod=*/(short)0, c, /*reuse_a=*/false, /*reuse_b=*/false);
}

// One K-chunk of a 16-row GEMV tile: W rows are the A operand (streamed),
// the vector is broadcast-loaded and value-masked into column 0 of B
// (v_cndmask, no divergent branch in the hot loop).
__device__ __forceinline__ void gemv_chunk(
    const float* __restrict__ wr,   // row base for this lane (incl. khalf offs)
    const float* __restrict__ vec,  // vector base for this lane (incl. khalf offs)
    int iters, bool col0,
    v8f& acc0, v8f& acc1, v8f& acc2, v8f& acc3) {
  const v2f zero = {0.f, 0.f};
  for (int it = 0; it < iters; ++it, wr += 16, vec += 16) {
    v2f a0 = *(const v2f*)(wr);
    v2f a1 = *(const v2f*)(wr + 4);
    v2f a2 = *(const v2f*)(wr + 8);
    v2f a3 = *(const v2f*)(wr + 12);
    v2f x0 = *(const v2f*)(vec);
    v2f x1 = *(const v2f*)(vec + 4);
    v2f x2 = *(const v2f*)(vec + 8);
    v2f x3 = *(const v2f*)(vec + 12);
    v2f b0 = col0 ? x0 : zero;
    v2f b1 = col0 ? x1 : zero;
    v2f b2 = col0 ? x2 : zero;
    v2f b3 = col0 ? x3 : zero;
    acc0 = wmma_step(acc0, a0, b0);
    acc1 = wmma_step(acc1, a1, b1);
    acc2 = wmma_step(acc2, a2, b2);
    acc3 = wmma_step(acc3, a3, b3);
  }
}

__global__ __launch_bounds__(256) void lstm_gates_wmma(
    const float* __restrict__ Wih, int Kin,
    const float* __restrict__ xin,
    const float* __restrict__ Whh,
    const float* __restrict__ hin,
    float* __restrict__ part) {
  const unsigned gtid  = blockIdx.x * blockDim.x + threadIdx.x;
  const unsigned wave  = gtid >> 5;          // wave32
  const unsigned lane  = threadIdx.x & 31u;
  const unsigned tile  = wave / SPLIT;       // 16-row tile index
  const unsigned s     = wave % SPLIT;       // K-split index
  const unsigned r0    = tile * 16u;
  if (r0 >= R_SZ) return;

  const unsigned mrow  = lane & 15u;   // M row inside the tile
  const unsigned khalf = lane >> 4;    // 0 -> K{0,1}, 1 -> K{2,3}
  const bool     col0  = (mrow == 0);  // lanes 0 & 16 feed B and own D column 0

  v8f acc0 = {}, acc1 = {}, acc2 = {}, acc3 = {};

  // ---- pass 1: W_ih (K = Kin) against xin ----
  {
    const int kc = Kin / SPLIT;
    const int kbeg = (int)s * kc;
    const float* wr = Wih + (size_t)(r0 + mrow) * (size_t)Kin + kbeg + 2u * khalf;
    const float* vp = xin + kbeg + 2u * khalf;
    gemv_chunk(wr, vp, kc / 16, col0, acc0, acc1, acc2, acc3);
  }

  // ---- pass 2: W_hh (K = H) against hin ----
  {
    const int kc = H_SZ / SPLIT;
    const int kbeg = (int)s * kc;
    const float* wr = Whh + (size_t)(r0 + mrow) * (size_t)H_SZ + kbeg + 2u * khalf;
    const float* vp = hin + kbeg + 2u * khalf;
    gemv_chunk(wr, vp, kc / 16, col0, acc0, acc1, acc2, acc3);
  }

  // D column 0: lane 0 holds rows r0..r0+7 (VGPR m), lane 16 rows r0+8..r0+15.
  if (col0) {
    const unsigned rb = r0 + khalf * 8u;
    float* dst = part + (size_t)s * R_SZ + rb;
#pragma unroll
    for (int m = 0; m < 8; ++m)
      dst[m] = acc0[m] + acc1[m] + acc2[m] + acc3[m];
  }
}

// Reduce split-K partials + biases, then the LSTM cell pointwise math.
// PyTorch gate order along the 4H axis: i, f, g, o.
__global__ __launch_bounds__(256) void lstm_pointwise(
    const float* __restrict__ part,
    const float* __restrict__ bih,
    const float* __restrict__ bhh,
    const float* __restrict__ cprev,
    float* __restrict__ hnew) {
  int j = blockIdx.x * blockDim.x + threadIdx.x;
  if (j >= H_SZ) return;
  float g[4];
#pragma unroll
  for (int q = 0; q < 4; ++q) {
    int r = j + q * H_SZ;
    float v = bih[r] + bhh[r];
#pragma unroll
    for (int s2 = 0; s2 < SPLIT; ++s2) v += part[(size_t)s2 * R_SZ + r];
    g[q] = v;
  }
  float ig = sigmoidf_(g[0]);
  float fg = sigmoidf_(g[1]);
  float gt = tanhf(g[2]);
  float og = sigmoidf_(g[3]);
  float cn = fg * cprev[j] + ig * gt;
  hnew[j] = og * tanhf(cn);
}

// Two 4096-length dot products + bias (+sigmoid for the D head).
__global__ __launch_bounds__(1024) void lstm_heads(
    const float* __restrict__ h2,
    const float* __restrict__ wL, const float* __restrict__ bL,
    const float* __restrict__ wD, const float* __restrict__ bD,
    float* __restrict__ out) {
  __shared__ float sL[1024];
  __shared__ float sD[1024];
  int t = threadIdx.x;
  float aL = 0.f, aD = 0.f;
  for (int j = t; j < H_SZ; j += 1024) {
    float h = h2[j];
    aL += h * wL[j];
    aD += h * wD[j];
  }
  sL[t] = aL;
  sD[t] = aD;
  __syncthreads();
  for (int s = 512; s > 0; s >>= 1) {
    if (t < s) {
      sL[t] += sL[t + s];
      sD[t] += sD[t + s];
    }
    __syncthreads();
  }
  if (t == 0) {
    out[1] = sL[0] + bL[0];                 // l_output
    out[0] = sigmoidf_(sD[0] + bD[0]);      // d_output
  }
}

extern "C" void kernel_launch(void* const* d_in, const int* in_sizes, int n_in,
                              void* d_out, int out_size, void* d_ws, size_t ws_size,
                              hipStream_t stream) {
  const float* data  = (const float*)d_in[0];
  const float* h0    = (const float*)d_in[1];   // [2,1,H]
  const float* c0    = (const float*)d_in[2];   // [2,1,H]
  const float* w_ih0 = (const float*)d_in[3];
  const float* w_hh0 = (const float*)d_in[4];
  const float* b_ih0 = (const float*)d_in[5];
  const float* b_hh0 = (const float*)d_in[6];
  const float* w_ih1 = (const float*)d_in[7];
  const float* w_hh1 = (const float*)d_in[8];
  const float* b_ih1 = (const float*)d_in[9];
  const float* b_hh1 = (const float*)d_in[10];
  const float* wL    = (const float*)d_in[11];
  const float* bL    = (const float*)d_in[12];
  const float* wD    = (const float*)d_in[13];
  const float* bD    = (const float*)d_in[14];
  float* out = (float*)d_out;

  // Workspace: split-K partials [SPLIT][R] (512KB), then h1[H], h2[H].
  float* part = (float*)d_ws;
  float* h1   = part + (size_t)SPLIT * R_SZ;
  float* h2   = h1 + H_SZ;

  // (R/16) tiles * SPLIT waves * 32 lanes / 256 threads-per-block
  const int gemv_blocks = (R_SZ / 16) * SPLIT * 32 / 256;  // 1024
  dim3 gblk(256);

  lstm_gates_wmma<<<gemv_blocks, gblk, 0, stream>>>(w_ih0, IN_SZ, data,
                                                    w_hh0, h0, part);
  lstm_pointwise<<<H_SZ / 256, 256, 0, stream>>>(part, b_ih0, b_hh0, c0, h1);

  lstm_gates_wmma<<<gemv_blocks, gblk, 0, stream>>>(w_ih1, H_SZ, h1,
                                                    w_hh1, h0 + H_SZ, part);
  lstm_pointwise<<<H_SZ / 256, 256, 0, stream>>>(part, b_ih1, b_hh1, c0 + H_SZ, h2);

  lstm_heads<<<1, 1024, 0, stream>>>(h2, wL, bL, wD, bD, out);
}